// eTofts_torch_38637525794947
// MI455X (gfx1250) — compile-verified
//
#include <hip/hip_runtime.h>
#include <stdint.h>

// ----- problem constants (from the reference) -----
#define NT       112                      // time points per row
#define ROWS     128                      // rows (batch elems) per block
#define LSTRIDE  116                      // padded LDS row stride (floats):
                                          //  - multiple of 4 -> rows 16B-aligned (b128 LDS ops legal)
                                          //  - 116 = 4*29 -> wave b128 reads hit every bank exactly 2x (optimal)

#define F_TR     0.005f
#define F_DELTT  (4.0f / 60.0f)
#define F_R1RLX  4.5f
#define F_COSA   0.96592582628906828675f  // cos(15 deg)
#define F_20SINA 5.17638090205041524807f  // 20 * sin(15 deg)

// One time-step of the extended-Tofts signal model.
// c is the IIR state: c_i = decay*c_{i-1} + CAp_i
// out = 20*sin(a)*(1-e) / (1 - e*cos(a)),  e = exp(-TR*(1/T10 + r1*ct))
__device__ __forceinline__ float etofts_step(float cap, float& c,
                                             float decay, float kd, float vpv,
                                             float K, float A)
{
    c = fmaf(c, decay, cap);                    // IIR state update
    float ct  = fmaf(vpv, cap, kd * c);         // vp*CAp + ktrans*dt*c
    float e   = __expf(fmaf(K, ct, A));         // exp(-TR/T10 - TR*r1*ct)
    float num = fmaf(-e, F_20SINA, F_20SINA);   // 20*sin(a)*(1-e)
    float den = fmaf(-e, F_COSA,   1.0f);       // 1 - e*cos(a)
    return num * __builtin_amdgcn_rcpf(den);
}

__launch_bounds__(ROWS)
__global__ void etofts_kernel(const float* __restrict__ ktrans,
                              const float* __restrict__ vp,
                              const float* __restrict__ ve,
                              const float* __restrict__ T10,
                              const float* __restrict__ CAp,
                              float* __restrict__ out)
{
    // 128 rows x 116 floats = 59,392 B -> 5 blocks (20 wave32) per 320KB WGP
    __shared__ __attribute__((aligned(16))) float tile[ROWS * LSTRIDE];

    const int tid = threadIdx.x;
    const long long base = (long long)blockIdx.x * (ROWS * NT);

    // ---------------------------------------------------------------
    // Stage in: CDNA5 async global->LDS DMA, B128 flavor (ASYNCcnt).
    // float4 index f4 = tid + 128*j -> consecutive lanes read consecutive
    // 16B chunks (fully coalesced 512B/wave requests). LDS float index
    // 4*(f4 + f4/28) lands each tile row at stride 116 (16B-aligned).
    // ---------------------------------------------------------------
    const float4* g4 = reinterpret_cast<const float4*>(CAp + base);
    #pragma unroll 4
    for (int j = 0; j < NT / 4; ++j) {          // 28 b128 DMAs per thread
        int f4 = tid + ROWS * j;
        unsigned ldsOff = (unsigned)(uintptr_t)(&tile[4 * (f4 + f4 / 28)]);
        const float4* ga = g4 + f4;
        asm volatile("global_load_async_to_lds_b128 %0, %1, off"
                     :: "v"(ldsOff), "v"(ga)
                     : "memory");
    }
    asm volatile("s_wait_asynccnt 0" ::: "memory");
    __syncthreads();

    // ---------------------------------------------------------------
    // Per-row scan: 28 bank-balanced ds_load_b128 from the padded row,
    // 4 model steps per load, direct b128 store of the result quad.
    // Strided stores write-combine in the 192MB L2 (each 128B line is
    // filled by 8 consecutive stores from one lane).
    // ---------------------------------------------------------------
    const int row    = blockIdx.x * ROWS + tid;
    const float kt   = ktrans[row];
    const float vpv  = vp[row];
    const float R10  = __builtin_amdgcn_rcpf(T10[row]);                    // 1/T10
    const float decay= __expf(-(kt * __builtin_amdgcn_rcpf(ve[row])) * F_DELTT);
    const float kd   = kt * F_DELTT;
    const float A    = -F_TR * R10;          // exp-argument offset
    const float K    = -F_TR * F_R1RLX;      // exp-argument scale on ct

    const float4* myrow = reinterpret_cast<const float4*>(&tile[tid * LSTRIDE]);
    float4* orow = reinterpret_cast<float4*>(out + (long long)row * NT);

    float c = 0.0f;
    #pragma unroll 4
    for (int k = 0; k < NT / 4; ++k) {
        float4 cap = myrow[k];
        float4 s;
        s.x = etofts_step(cap.x, c, decay, kd, vpv, K, A);
        s.y = etofts_step(cap.y, c, decay, kd, vpv, K, A);
        s.z = etofts_step(cap.z, c, decay, kd, vpv, K, A);
        s.w = etofts_step(cap.w, c, decay, kd, vpv, K, A);
        orow[k] = s;
    }
}

extern "C" void kernel_launch(void* const* d_in, const int* in_sizes, int n_in,
                              void* d_out, int out_size, void* d_ws, size_t ws_size,
                              hipStream_t stream)
{
    (void)n_in; (void)out_size; (void)d_ws; (void)ws_size;

    const float* ktrans = (const float*)d_in[0];
    const float* vp     = (const float*)d_in[1];
    const float* ve     = (const float*)d_in[2];
    const float* T10    = (const float*)d_in[3];
    const float* CAp    = (const float*)d_in[4];
    float* out          = (float*)d_out;

    const int batch = in_sizes[0];           // 262144
    const int grid  = batch / ROWS;          // 2048 blocks of 128 threads

    etofts_kernel<<<grid, ROWS, 0, stream>>>(ktrans, vp, ve, T10, CAp, out);
}